// Aggregator_14817637171432
// MI455X (gfx1250) — compile-verified
//
#include <hip/hip_runtime.h>

#define N_NODES 100000
#define N_EDGES 3200000
#define N_REL   32
#define DIM     64
#define N_TILES (N_NODES / 16)   // 6250 exactly

typedef __attribute__((ext_vector_type(2))) float v2f;
typedef __attribute__((ext_vector_type(8))) float v8f;

// Device-scope, no-return f32 atomic add: fire-and-forget RMW at the L2
// (coherent point for DEV scope). No pre-op return -> tracked on STOREcnt,
// implicit s_wait_idle before s_endpgm covers completion.
__device__ __forceinline__ void atomic_add_f32_dev(float* p, float v) {
    asm volatile("global_atomic_add_f32 %0, %1, off scope:SCOPE_DEV"
                 :: "v"(p), "v"(v) : "memory");
}

// ---------------------------------------------------------------------------
// Kernel 1: zero the side accumulator (float4 stores, exact grid coverage)
// ---------------------------------------------------------------------------
__global__ __launch_bounds__(256) void zero_side(float4* __restrict__ side4) {
    int i = blockIdx.x * 256 + threadIdx.x;   // 6250 blocks * 256 = 1.6M float4 exactly
    float4 z = {0.f, 0.f, 0.f, 0.f};
    side4[i] = z;
}

// ---------------------------------------------------------------------------
// Kernel 2: fused COO scatter for both SpMMs.
// 16 lanes per edge, each lane owns a float4 column chunk (16*4 = 64 = DIM).
// side[row] += vals*ego[col]  and  side[row_r] += 0.1*vals_r*rel[rel_idx]
// ---------------------------------------------------------------------------
__global__ __launch_bounds__(256) void scatter_edges(
    const float* __restrict__ ego, const float* __restrict__ rel,
    const int*  __restrict__ row_idx, const int* __restrict__ col_idx,
    const float* __restrict__ vals,
    const int*  __restrict__ row_idx_r, const int* __restrict__ rel_idx,
    const float* __restrict__ vals_r,
    float* __restrict__ side)
{
    long t = (long)blockIdx.x * 256 + threadIdx.x;
    int e = (int)(t >> 4);
    int c = ((int)t & 15) * 4;
    if (e >= N_EDGES) return;

    { // SpMM 1: ego gather (25.6MB table, L2-resident)
        int   r   = row_idx[e];
        int   col = col_idx[e];
        float v   = vals[e];
        const float4 x = *(const float4*)(ego + (long)col * DIM + c);
        float* d = side + (long)r * DIM + c;
        atomic_add_f32_dev(d + 0, v * x.x);
        atomic_add_f32_dev(d + 1, v * x.y);
        atomic_add_f32_dev(d + 2, v * x.z);
        atomic_add_f32_dev(d + 3, v * x.w);
    }
    { // SpMM 2: rel gather (8KB table, cache-hot), pre-scaled by 0.1
        int   r  = row_idx_r[e];
        int   ri = rel_idx[e];
        float v  = 0.1f * vals_r[e];
        const float4 x = *(const float4*)(rel + ri * DIM + c);
        float* d = side + (long)r * DIM + c;
        atomic_add_f32_dev(d + 0, v * x.x);
        atomic_add_f32_dev(d + 1, v * x.y);
        atomic_add_f32_dev(d + 2, v * x.z);
        atomic_add_f32_dev(d + 3, v * x.w);
    }
}

// ---------------------------------------------------------------------------
// Kernel 3: fused dual GEMM + bias + leaky_relu + add, via WMMA f32 16x16x4.
// One wave per 16-row tile; 4 N-tiles of 16 cover DIM=64.
//   sum = lrelu((ego+side) @ W1.T + b1); bi = lrelu((ego*side) @ W2.T + b2)
//   out = sum + bi
// A layout (f32 16x4): lane&15 = M row, lane>>4 picks K pair {k0,k0+1}/{k0+2,k0+3}
//   -> contiguous float2 per lane. B layout mirrors with lane&15 = N.
// ---------------------------------------------------------------------------
__global__ __launch_bounds__(256) void dense_wmma(
    const float* __restrict__ ego, const float* __restrict__ side,
    const float* __restrict__ W1, const float* __restrict__ b1,
    const float* __restrict__ W2, const float* __restrict__ b2,
    float* __restrict__ out)
{
    const int lane = threadIdx.x & 31;
    const int wave = threadIdx.x >> 5;
    const int tile = blockIdx.x * 8 + wave;
    if (tile >= N_TILES) return;          // wave-uniform: EXEC all-ones inside

    const int m0 = tile * 16;
    const int h  = lane >> 4;             // lane half selects K pair
    const int l  = lane & 15;
    const int m  = m0 + l;                // A row for this lane

    v8f accS[4] = { {0,0,0,0,0,0,0,0}, {0,0,0,0,0,0,0,0},
                    {0,0,0,0,0,0,0,0}, {0,0,0,0,0,0,0,0} };
    v8f accB[4] = { {0,0,0,0,0,0,0,0}, {0,0,0,0,0,0,0,0},
                    {0,0,0,0,0,0,0,0}, {0,0,0,0,0,0,0,0} };

    for (int k0 = 0; k0 < DIM; k0 += 4) {
        const int kk = k0 + 2 * h;
        v2f eg = *(const v2f*)(ego  + (long)m * DIM + kk);
        v2f sd = *(const v2f*)(side + (long)m * DIM + kk);
        v2f aS = eg + sd;                 // add_emb operand
        v2f aB = eg * sd;                 // ego*side operand
        #pragma unroll
        for (int nt = 0; nt < 4; nt++) {
            const int n = nt * 16 + l;    // B column this lane supplies
            v2f bS = *(const v2f*)(W1 + n * DIM + kk);  // B[k][n] = W1[n][k]
            v2f bB = *(const v2f*)(W2 + n * DIM + kk);
            accS[nt] = __builtin_amdgcn_wmma_f32_16x16x4_f32(
                false, aS, false, bS, (short)0, accS[nt], false, false);
            accB[nt] = __builtin_amdgcn_wmma_f32_16x16x4_f32(
                false, aB, false, bB, (short)0, accB[nt], false, false);
        }
    }

    // Epilogue: C/D layout -> VGPR j holds row j (lanes 0-15) / j+8 (lanes 16-31)
    #pragma unroll
    for (int nt = 0; nt < 4; nt++) {
        const int n = nt * 16 + l;
        const float bb1 = b1[n];
        const float bb2 = b2[n];
        #pragma unroll
        for (int j = 0; j < 8; j++) {
            float s  = accS[nt][j] + bb1;
            s  = (s  > 0.f) ? s  : 0.01f * s;
            float bi = accB[nt][j] + bb2;
            bi = (bi > 0.f) ? bi : 0.01f * bi;
            out[(long)(m0 + j + 8 * h) * DIM + n] = s + bi;
        }
    }
}

// ---------------------------------------------------------------------------
extern "C" void kernel_launch(void* const* d_in, const int* in_sizes, int n_in,
                              void* d_out, int out_size, void* d_ws, size_t ws_size,
                              hipStream_t stream)
{
    const float* ego       = (const float*)d_in[0];
    const float* rel       = (const float*)d_in[1];
    const int*   row_idx   = (const int*)  d_in[2];
    const int*   col_idx   = (const int*)  d_in[3];
    const float* vals      = (const float*)d_in[4];
    const int*   row_idx_r = (const int*)  d_in[5];
    const int*   rel_idx   = (const int*)  d_in[6];
    const float* vals_r    = (const float*)d_in[7];
    const float* W1        = (const float*)d_in[8];
    const float* b1        = (const float*)d_in[9];
    const float* W2        = (const float*)d_in[10];
    const float* b2        = (const float*)d_in[11];

    float* out  = (float*)d_out;
    float* side = (float*)d_ws;           // 100000*64*4 = 25.6 MB scratch

    // 1) zero accumulator: 6.4M floats = 1.6M float4 / 256 = 6250 blocks exact
    zero_side<<<6250, 256, 0, stream>>>((float4*)side);

    // 2) scatter both SpMMs: 3.2M edges * 16 lanes / 256 = 200000 blocks exact
    scatter_edges<<<200000, 256, 0, stream>>>(ego, rel, row_idx, col_idx, vals,
                                              row_idx_r, rel_idx, vals_r, side);

    // 3) fused WMMA GEMMs + activation: 6250 tiles, 8 waves/block
    dense_wmma<<<(N_TILES + 7) / 8, 256, 0, stream>>>(ego, side, W1, b1, W2, b2, out);
}